// GNNLayer_6846177869955
// MI455X (gfx1250) — compile-verified
//
#include <hip/hip_runtime.h>
#include <hip/hip_bf16.h>

// ---------------------------------------------------------------------------
// GNN interaction-network layer stack for MI455X (gfx1250, wave32, WMMA).
//
//  * fp32 matrix math on the matrix pipes via V_WMMA_F32_16X16X4_F32
//    (exact vs. the fp32 reference; workload is gather/L2-bound).
//  * Weight-stationary persistent workgroups: all three MLP weight matrices
//    of a layer live in the 320KB/WGP LDS; blocks grid-stride over tiles.
//  * Edge kernel processes 32 edges/tile as TWO 16x16 M-tiles with two
//    independent WMMA accumulators sharing each B fragment: breaks the
//    serial D->C dependence chain seen in round-1 asm, halves B LDS traffic
//    and barrier overhead. Stage-2 output aliases the (dead) A-staging
//    buffer to stay under the 320KB LDS limit.
//  * Edge MLP + bias + relu + scatter-add (segment_sum) fused; messages
//    never touch global memory. Node MLP + residual + LayerNorm fused.
// ---------------------------------------------------------------------------

typedef float v2f __attribute__((ext_vector_type(2)));
typedef float v8f __attribute__((ext_vector_type(8)));

constexpr int HD  = 128;   // hidden dim
constexpr int HP  = 132;   // padded LDS stride for [K][128] tiles (bank-conflict-free)
constexpr int TEE = 32;    // edges per tile (two 16-row WMMA M-tiles)
constexpr int TN  = 16;    // nodes per tile
constexpr int KE  = 264;   // edge-MLP K (263 padded to multiple of 4)
constexpr int KEP = 268;   // padded LDS stride for edge A tile
constexpr int KN  = 256;   // node-MLP K
constexpr int KNP = 260;   // padded LDS stride for node A tile
constexpr int LAYERS = 4;

// Edge LDS: W1(264x132) + W2,W3(128x132) + As(32x268) + M1(32x132) + csh(32 int)
// M2 aliases As (As is dead after stage 1).
constexpr size_t EDGE_LDS_BYTES =
    ((size_t)KE * HP + 2 * (size_t)HD * HP + (size_t)TEE * KEP + (size_t)TEE * HP) * 4
    + TEE * sizeof(int);                                   // = 325,888 B <= 320KB limit
constexpr size_t NODE_LDS_BYTES =
    ((size_t)KN * HP + 2 * (size_t)HD * HP + (size_t)TN * KNP + 2 * (size_t)TN * HP) * 4;

// ---------------------------------------------------------------------------
// WMMA helpers: D += A(16xK) * B(Kx16) in K-steps of 4 (V_WMMA_F32_16X16X4_F32)
//  A frag (16x4): lanes 0-15 -> M=lane, K={k0,k0+1}; lanes 16-31 -> K={k0+2,k0+3}
//  B frag (4x16): lanes 0-15 -> rows {k0,k0+1}, N=lane; lanes 16-31 -> rows {k0+2,k0+3}
// ---------------------------------------------------------------------------

__device__ __forceinline__ v8f mma_span_f32(const float* __restrict__ A, int aStride,
                                            const float* __restrict__ Bcol, int bStride,
                                            int kTot, int lane, v8f acc)
{
    const int mA   = lane & 15;
    const int kOff = (lane >> 4) << 1;          // 0 or 2
    const float* aRow = A + mA * aStride + kOff;
    const float* bPtr = Bcol + kOff * bStride;  // Bcol pre-offset by this lane's column
    for (int k0 = 0; k0 < kTot; k0 += 4) {
        v2f a, b;
        a.x = aRow[k0];
        a.y = aRow[k0 + 1];
        b.x = bPtr[(k0    ) * bStride];
        b.y = bPtr[(k0 + 1) * bStride];
        acc = __builtin_amdgcn_wmma_f32_16x16x4_f32(false, a, false, b,
                                                    (short)0, acc, false, false);
    }
    return acc;
}

// Two M-tiles sharing each B fragment: two independent accumulate chains.
__device__ __forceinline__ void mma_span_f32_x2(const float* __restrict__ A0,
                                                const float* __restrict__ A1, int aStride,
                                                const float* __restrict__ Bcol, int bStride,
                                                int kTot, int lane, v8f& acc0, v8f& acc1)
{
    const int mA   = lane & 15;
    const int kOff = (lane >> 4) << 1;
    const float* aRow0 = A0 + mA * aStride + kOff;
    const float* aRow1 = A1 + mA * aStride + kOff;
    const float* bPtr  = Bcol + kOff * bStride;
    for (int k0 = 0; k0 < kTot; k0 += 4) {
        v2f a0, a1, b;
        b.x  = bPtr[(k0    ) * bStride];
        b.y  = bPtr[(k0 + 1) * bStride];
        a0.x = aRow0[k0];  a0.y = aRow0[k0 + 1];
        a1.x = aRow1[k0];  a1.y = aRow1[k0 + 1];
        acc0 = __builtin_amdgcn_wmma_f32_16x16x4_f32(false, a0, false, b,
                                                     (short)0, acc0, false, false);
        acc1 = __builtin_amdgcn_wmma_f32_16x16x4_f32(false, a1, false, b,
                                                     (short)0, acc1, false, false);
    }
}

// ---------------------------------------------------------------------------
// Small setup kernels
// ---------------------------------------------------------------------------

__global__ void geom_kernel(const float* __restrict__ pos,
                            const int* __restrict__ rowi,
                            const int* __restrict__ coli,
                            float* __restrict__ geom, int E_)
{
    int e = blockIdx.x * blockDim.x + threadIdx.x;
    if (e >= E_) return;
    int r = rowi[e], c = coli[e];
    float dx = pos[3 * c + 0] - pos[3 * r + 0];
    float dy = pos[3 * c + 1] - pos[3 * r + 1];
    float dz = pos[3 * c + 2] - pos[3 * r + 2];
    float d  = sqrtf(dx * dx + dy * dy + dz * dz);
    geom[4 * e + 0] = dx; geom[4 * e + 1] = dy;
    geom[4 * e + 2] = dz; geom[4 * e + 3] = d;
}

__global__ void embed_kernel(const float* __restrict__ x,
                             const float* __restrict__ W,
                             const float* __restrict__ b,
                             float* __restrict__ h, long total)
{
    long idx = (long)blockIdx.x * blockDim.x + threadIdx.x;
    if (idx >= total) return;
    int  j = (int)(idx & (HD - 1));
    long i = idx >> 7;
    float acc = b[j];
#pragma unroll
    for (int k = 0; k < 16; ++k) acc = fmaf(x[i * 16 + k], W[k * HD + j], acc);
    h[idx] = acc;
}

__global__ void zero_kernel(float* __restrict__ p, long total)
{
    long idx = (long)blockIdx.x * blockDim.x + threadIdx.x;
    if (idx < total) p[idx] = 0.f;
}

// ---------------------------------------------------------------------------
// Fused edge MLP + segment-sum scatter, 32 edges per tile.
// A row per edge: [h[col](128) | h[row](128) | rel_pos(3),dist(1),rel_mom(3) | pad]
// ---------------------------------------------------------------------------
__global__ void __launch_bounds__(256, 1)
edge_kernel(const float* __restrict__ hin,
            const float* __restrict__ geom,
            const int* __restrict__ rowi,
            const int* __restrict__ coli,
            const float* __restrict__ We1, const float* __restrict__ be1,
            const float* __restrict__ We2, const float* __restrict__ be2,
            const float* __restrict__ We3, const float* __restrict__ be3,
            float* __restrict__ agg, int E_)
{
    extern __shared__ float smem[];
    float* W1s = smem;                  // KE x HP  (row 263 zero-padded)
    float* W2s = W1s + KE * HP;         // HD x HP
    float* W3s = W2s + HD * HP;         // HD x HP
    float* As  = W3s + HD * HP;         // TEE x KEP  (stage-1 A)
    float* M1s = As  + TEE * KEP;       // TEE x HP
    float* M2s = As;                    // TEE x HP, ALIASES As (dead after stage 1)
    int*   csh = (int*)(M1s + TEE * HP);

    const int tid = threadIdx.x;

    // Stage weights into LDS once (weight-stationary across all tiles).
    for (int i = tid; i < KE * HD; i += 256) {
        int kk = i >> 7, j = i & (HD - 1);
        W1s[kk * HP + j] = (kk < 263) ? We1[i] : 0.f;
    }
    for (int i = tid; i < HD * HD; i += 256) {
        int kk = i >> 7, j = i & (HD - 1);
        W2s[kk * HP + j] = We2[i];
        W3s[kk * HP + j] = We3[i];
    }
    __syncthreads();

    const int lane = tid & 31;
    const int wav  = tid >> 5;                 // 8 waves: one 16-col slice each
    const int nCol = (wav << 4) + (lane & 15);
    const int mHi  = lane >> 4;
    const float b1 = be1[nCol];
    const float b2 = be2[nCol];
    const float b3 = be3[nCol];

    const int numTiles = (E_ + TEE - 1) / TEE;
    for (int t = blockIdx.x; t < numTiles; t += gridDim.x) {
        const int e0 = t * TEE;
        if (tid < TEE) csh[tid] = (e0 + tid < E_) ? coli[e0 + tid] : -1;

        // Gather/stage the 32x263 edge-input tile.
        for (int el = 0; el < TEE; ++el) {
            const int  e     = e0 + el;
            const bool valid = (e < E_);
            const int  c     = valid ? coli[e] : 0;
            const int  r     = valid ? rowi[e] : 0;
            float* dst = As + el * KEP;
            for (int k = tid; k < KE; k += 256) {
                float v = 0.f;
                if (valid) {
                    if      (k < HD)          v = hin[(long)c * HD + k];
                    else if (k < 2 * HD)      v = hin[(long)r * HD + (k - HD)];
                    else if (k < 2 * HD + 4)  v = geom[(long)e * 4 + (k - 2 * HD)];
                    else if (k < 263)         v = hin[(long)c * HD + 3 + (k - 260)]
                                                - hin[(long)r * HD + 3 + (k - 260)];
                }
                dst[k] = v;
            }
        }
        __syncthreads();

        // m1 = relu(A @ We1 + b1)   (two M-tiles, shared B fragments)
        v8f p0 = {}, p1 = {};
        mma_span_f32_x2(As, As + 16 * KEP, KEP, W1s + nCol, HP, KE, lane, p0, p1);
#pragma unroll
        for (int rr = 0; rr < 8; ++rr) {
            const int mrow = rr + (mHi << 3);
            M1s[mrow * HP + nCol]        = fmaxf(p0[rr] + b1, 0.f);
            M1s[(mrow + 16) * HP + nCol] = fmaxf(p1[rr] + b1, 0.f);
        }
        __syncthreads();   // As is dead past this point -> M2s may overwrite it

        // m2 = relu(m1 @ We2 + b2)
        v8f q0 = {}, q1 = {};
        mma_span_f32_x2(M1s, M1s + 16 * HP, HP, W2s + nCol, HP, HD, lane, q0, q1);
        __syncthreads();   // all waves done READING M2s-aliased space? (no reads yet)
#pragma unroll
        for (int rr = 0; rr < 8; ++rr) {
            const int mrow = rr + (mHi << 3);
            M2s[mrow * HP + nCol]        = fmaxf(q0[rr] + b2, 0.f);
            M2s[(mrow + 16) * HP + nCol] = fmaxf(q1[rr] + b2, 0.f);
        }
        __syncthreads();

        // m3 = m2 @ We3 + b3, scatter-add to agg[col]
        v8f r0 = {}, r1 = {};
        mma_span_f32_x2(M2s, M2s + 16 * HP, HP, W3s + nCol, HP, HD, lane, r0, r1);
#pragma unroll
        for (int rr = 0; rr < 8; ++rr) {
            const int mrow = rr + (mHi << 3);
            const int c0 = csh[mrow];
            const int c1 = csh[mrow + 16];
            if (c0 >= 0) atomicAdd(agg + (long)c0 * HD + nCol, r0[rr] + b3);
            if (c1 >= 0) atomicAdd(agg + (long)c1 * HD + nCol, r1[rr] + b3);
        }
        __syncthreads();
    }
}

// ---------------------------------------------------------------------------
// Fused node MLP + residual + LayerNorm (16 nodes per tile).
// ---------------------------------------------------------------------------
__global__ void __launch_bounds__(256, 1)
node_kernel(const float* __restrict__ hin,
            const float* __restrict__ aggp,
            const float* __restrict__ Wn1, const float* __restrict__ bn1,
            const float* __restrict__ Wn2, const float* __restrict__ bn2,
            const float* __restrict__ Wn3, const float* __restrict__ bn3,
            const float* __restrict__ lng, const float* __restrict__ lnb,
            float* __restrict__ hout, int N_)
{
    extern __shared__ float smem[];
    float* W1s = smem;                 // KN x HP
    float* W2s = W1s + KN * HP;        // HD x HP
    float* W3s = W2s + HD * HP;        // HD x HP
    float* As  = W3s + HD * HP;        // TN x KNP   (keeps original h for residual)
    float* U1s = As  + TN * KNP;       // TN x HP
    float* U2s = U1s + TN * HP;        // TN x HP

    const int tid = threadIdx.x;
    for (int i = tid; i < KN * HD; i += 256)
        W1s[(i >> 7) * HP + (i & (HD - 1))] = Wn1[i];
    for (int i = tid; i < HD * HD; i += 256) {
        W2s[(i >> 7) * HP + (i & (HD - 1))] = Wn2[i];
        W3s[(i >> 7) * HP + (i & (HD - 1))] = Wn3[i];
    }
    __syncthreads();

    const int lane = tid & 31;
    const int wav  = tid >> 5;
    const int nCol = (wav << 4) + (lane & 15);
    const int mHi  = lane >> 4;
    const float b1 = bn1[nCol], b2 = bn2[nCol], b3 = bn3[nCol];

    const int numTiles = (N_ + TN - 1) / TN;
    for (int t = blockIdx.x; t < numTiles; t += gridDim.x) {
        const int n0 = t * TN;

        // Stage A = [h | agg] for 16 nodes (256 threads cover exactly K=256).
        for (int m = 0; m < TN; ++m) {
            const int  node  = n0 + m;
            const bool valid = node < N_;
            const int  k     = tid;
            float v = 0.f;
            if (valid)
                v = (k < HD) ? hin[(long)node * HD + k]
                             : aggp[(long)node * HD + (k - HD)];
            As[m * KNP + k] = v;
        }
        __syncthreads();

        v8f a1 = {};
        a1 = mma_span_f32(As, KNP, W1s + nCol, HP, KN, lane, a1);
#pragma unroll
        for (int rr = 0; rr < 8; ++rr)
            U1s[(rr + (mHi << 3)) * HP + nCol] = fmaxf(a1[rr] + b1, 0.f);
        __syncthreads();

        v8f a2 = {};
        a2 = mma_span_f32(U1s, HP, W2s + nCol, HP, HD, lane, a2);
#pragma unroll
        for (int rr = 0; rr < 8; ++rr)
            U2s[(rr + (mHi << 3)) * HP + nCol] = fmaxf(a2[rr] + b2, 0.f);
        __syncthreads();

        // u = m2 @ Wn3 + b3 ; pre-LN = h + u  (h still resident in As[:, :128])
        v8f a3 = {};
        a3 = mma_span_f32(U2s, HP, W3s + nCol, HP, HD, lane, a3);
#pragma unroll
        for (int rr = 0; rr < 8; ++rr) {
            const int mrow = rr + (mHi << 3);
            const float resid = As[mrow * KNP + nCol];
            U1s[mrow * HP + nCol] = a3[rr] + b3 + resid;
        }
        __syncthreads();

        // LayerNorm: 8 lanes per row, wave32 shuffle reduction.
        if (tid < HD) {
            const int m = tid >> 3, sub = tid & 7;
            const int node = n0 + m;
            if (node < N_) {
                const float* trow = U1s + m * HP;
                float s = 0.f;
                for (int j = sub; j < HD; j += 8) s += trow[j];
                s += __shfl_xor(s, 1); s += __shfl_xor(s, 2); s += __shfl_xor(s, 4);
                const float mean = s * (1.f / HD);
                float vv = 0.f;
                for (int j = sub; j < HD; j += 8) { float d = trow[j] - mean; vv += d * d; }
                vv += __shfl_xor(vv, 1); vv += __shfl_xor(vv, 2); vv += __shfl_xor(vv, 4);
                const float inv = rsqrtf(vv * (1.f / HD) + 1e-5f);
                for (int j = sub; j < HD; j += 8)
                    hout[(long)node * HD + j] = (trow[j] - mean) * inv * lng[j] + lnb[j];
            }
        }
        __syncthreads();
    }
}

// ---------------------------------------------------------------------------

extern "C" void kernel_launch(void* const* d_in, const int* in_sizes, int n_in,
                              void* d_out, int out_size, void* d_ws, size_t ws_size,
                              hipStream_t stream)
{
    (void)n_in; (void)out_size; (void)ws_size;

    const float* x    = (const float*)d_in[0];
    const float* pos  = (const float*)d_in[1];
    const int*   eidx = (const int*)  d_in[2];
    const float* W_in = (const float*)d_in[3];
    const float* b_in = (const float*)d_in[4];
    const float* We1  = (const float*)d_in[5];
    const float* be1  = (const float*)d_in[6];
    const float* We2  = (const float*)d_in[7];
    const float* be2  = (const float*)d_in[8];
    const float* We3  = (const float*)d_in[9];
    const float* be3  = (const float*)d_in[10];
    const float* Wn1  = (const float*)d_in[11];
    const float* bn1  = (const float*)d_in[12];
    const float* Wn2  = (const float*)d_in[13];
    const float* bn2  = (const float*)d_in[14];
    const float* Wn3  = (const float*)d_in[15];
    const float* bn3  = (const float*)d_in[16];
    const float* ln_g = (const float*)d_in[17];
    const float* ln_b = (const float*)d_in[18];

    const int N_ = in_sizes[0] / 16;   // 50000
    const int E_ = in_sizes[2] / 2;    // 800000
    const int* rowp = eidx;
    const int* colp = eidx + E_;

    float* h    = (float*)d_ws;                      // N x 128
    float* agg  = h   + (size_t)N_ * HD;             // N x 128
    float* geom = agg + (size_t)N_ * HD;             // E x 4

    const long NH = (long)N_ * HD;

    geom_kernel<<<(E_ + 255) / 256, 256, 0, stream>>>(pos, rowp, colp, geom, E_);
    embed_kernel<<<(int)((NH + 255) / 256), 256, 0, stream>>>(x, W_in, b_in, h, NH);

    const int persistentBlocks = 1024;  // grid-stride; 1 WG/WGP due to LDS footprint

    for (int l = 0; l < LAYERS; ++l) {
        zero_kernel<<<(int)((NH + 255) / 256), 256, 0, stream>>>(agg, NH);

        edge_kernel<<<persistentBlocks, 256, (unsigned)EDGE_LDS_BYTES, stream>>>(
            h, geom, rowp, colp,
            We1 + (size_t)l * 263 * HD, be1 + l * HD,
            We2 + (size_t)l * HD * HD,  be2 + l * HD,
            We3 + (size_t)l * HD * HD,  be3 + l * HD,
            agg, E_);

        float* hout = (l == LAYERS - 1) ? (float*)d_out : h;
        node_kernel<<<persistentBlocks, 256, (unsigned)NODE_LDS_BYTES, stream>>>(
            h, agg,
            Wn1 + (size_t)l * 256 * HD, bn1 + l * HD,
            Wn2 + (size_t)l * HD * HD,  bn2 + l * HD,
            Wn3 + (size_t)l * HD * HD,  bn3 + l * HD,
            ln_g + l * HD, ln_b + l * HD,
            hout, N_);
    }
}